// PlaneSweepCostVolume_84387517432235
// MI455X (gfx1250) — compile-verified
//
#include <hip/hip_runtime.h>
#include <hip/hip_bf16.h>
#include <math.h>

#define NUM_D   128
#define H_      240
#define W_      320
#define C_      64
#define VS_     4
#define HW_     (H_ * W_)
#define DEPTH_MIN_ 0.5f
#define DEPTH_MAX_ 15.0f

typedef __attribute__((ext_vector_type(2))) float v2f;
typedef __attribute__((ext_vector_type(8))) float v8f;

// ---------------- small 3x3 helpers ----------------
__device__ __forceinline__ void mat3_mul(const float* A, const float* B, float* Co) {
    for (int i = 0; i < 3; ++i)
        for (int j = 0; j < 3; ++j)
            Co[i * 3 + j] = A[i * 3 + 0] * B[0 * 3 + j] +
                            A[i * 3 + 1] * B[1 * 3 + j] +
                            A[i * 3 + 2] * B[2 * 3 + j];
}
__device__ __forceinline__ void mat3_vec(const float* A, const float* x, float* y) {
    for (int i = 0; i < 3; ++i)
        y[i] = A[i * 3 + 0] * x[0] + A[i * 3 + 1] * x[1] + A[i * 3 + 2] * x[2];
}
__device__ __forceinline__ void mat3_inv(const float* m, float* o) {
    float a = m[0], b = m[1], c = m[2];
    float d = m[3], e = m[4], f = m[5];
    float g = m[6], h = m[7], i = m[8];
    float A0 =  (e * i - f * h);
    float A1 = -(d * i - f * g);
    float A2 =  (d * h - e * g);
    float det = a * A0 + b * A1 + c * A2;
    float id = 1.0f / det;   // host-side-style setup math; once per launch, precision ok
    o[0] = A0 * id;             o[1] = -(b * i - c * h) * id;  o[2] =  (b * f - c * e) * id;
    o[3] = A1 * id;             o[4] =  (a * i - c * g) * id;  o[5] = -(a * f - c * d) * id;
    o[6] = A2 * id;             o[7] = -(a * h - b * g) * id;  o[8] =  (a * e - b * d) * id;
}

// ws layout (floats): [0..35] P[v][9], [36..47] bvec[v][3], [64..64+HW) rinv_ref
__global__ void setup_kernel(const float* __restrict__ Kref,
                             const float* __restrict__ c2w_ref,
                             const float* __restrict__ Ksrcs,
                             const float* __restrict__ c2w_srcs,
                             float* __restrict__ ws,
                             float* __restrict__ depth_out) {
    int t = threadIdx.x;
    if (t < NUM_D) {
        float inv0 = 1.0f / DEPTH_MAX_;
        float inv1 = 1.0f / DEPTH_MIN_;
        float step = (inv1 - inv0) / (float)(NUM_D - 1);
        depth_out[t] = 1.0f / fmaf((float)t, step, inv0);
    }
    if (t < VS_) {
        float Kinv[9]; mat3_inv(Kref, Kinv);
        float Rr[9], tr[3];
        for (int i = 0; i < 3; ++i) {
            for (int j = 0; j < 3; ++j) Rr[i * 3 + j] = c2w_ref[i * 4 + j];
            tr[i] = c2w_ref[i * 4 + 3];
        }
        const float* Ms = c2w_srcs + t * 16;
        float Rw[9], ts[3];
        for (int i = 0; i < 3; ++i) {
            for (int j = 0; j < 3; ++j) Rw[i * 3 + j] = Ms[j * 4 + i];  // R_src^T
            ts[i] = Ms[i * 4 + 3];
        }
        const float* Ks = Ksrcs + t * 9;
        float T1[9], M3[9], P[9];
        mat3_mul(Rw, Rr, T1);     // Rw2c * R_ref
        mat3_mul(Ks, T1, M3);     // K_src * Rw2c * R_ref
        mat3_mul(M3, Kinv, P);    // fold K_ref^-1:  A(pix) = P * (x,y,1)
        float dt[3] = { tr[0] - ts[0], tr[1] - ts[1], tr[2] - ts[2] };
        float u[3], bv[3];
        mat3_vec(Rw, dt, u);
        mat3_vec(Ks, u, bv);      // bvec = K_src * Rw2c * (t_ref - t_src)
        for (int i = 0; i < 9; ++i) ws[t * 9 + i] = P[i];
        for (int i = 0; i < 3; ++i) ws[36 + t * 3 + i] = bv[i];
    }
}

__global__ void refnorm_kernel(const float* __restrict__ ref, float* __restrict__ rinv) {
    int p = blockIdx.x * blockDim.x + threadIdx.x;
    if (p >= HW_) return;
    float s = 0.0f;
    for (int c = 0; c < C_; ++c) {
        float v = ref[(size_t)c * HW_ + p];
        s = fmaf(v, v, s);
    }
    float n = __builtin_amdgcn_sqrtf(s);
    rinv[p] = __builtin_amdgcn_rcpf(fmaxf(n, 1e-12f));
}

// Wave mapping: lane L -> pixel (group_base + L%16), views {0,1} for L<16, {2,3} for L>=16.
// The 4-view mean is the K=4 contraction D = A(16x4, all 0.25) x B(4x16, sims);
// B lane layout: lane = column (pixel), VGPR j = K = 2*(lane/16)+j  -> exactly our sims.
// D VGPR0: lane L holds mean of pixel L%16 (all rows identical since A rows identical).
__global__ __launch_bounds__(128) void cost_kernel(const float* __restrict__ ref,
                                                   const float* __restrict__ src,
                                                   const float* __restrict__ ws,
                                                   const float* __restrict__ rinvbuf,
                                                   float* __restrict__ out) {
    const int lane = threadIdx.x & 31;
    const int wave = threadIdx.x >> 5;
    const int pix  = blockIdx.x * 64 + wave * 16 + (lane & 15);
    const int d    = blockIdx.y;
    const int px   = pix % W_;
    const int py   = pix / W_;
    const int vbase = (lane >> 4) * 2;

    const float inv0 = 1.0f / DEPTH_MAX_;
    const float inv1 = 1.0f / DEPTH_MIN_;
    const float step = (inv1 - inv0) / (float)(NUM_D - 1);
    const float depth = __builtin_amdgcn_rcpf(fmaf((float)d, step, inv0));

    const float xf = (float)px;
    const float yf = (float)py;
    const float rref = rinvbuf[pix];

    int   o00[2], o01[2], o10[2], o11[2];
    float w00[2], w01[2], w10[2], w11[2];
    bool  vld[2];
    const float* basep[2];

#pragma unroll
    for (int j = 0; j < 2; ++j) {
        const int v = vbase + j;
        const float* P  = ws + v * 9;
        const float* bv = ws + 36 + v * 3;
        float A0 = fmaf(P[0], xf, fmaf(P[1], yf, P[2]));
        float A1 = fmaf(P[3], xf, fmaf(P[4], yf, P[5]));
        float A2 = fmaf(P[6], xf, fmaf(P[7], yf, P[8]));
        float p0 = fmaf(depth, A0, bv[0]);
        float p1 = fmaf(depth, A1, bv[1]);
        float z  = fmaf(depth, A2, bv[2]);
        float zs = (fabsf(z) > 1e-6f) ? z : 1e-6f;
        float rz = __builtin_amdgcn_rcpf(zs);   // v_rcp_f32: no EXEC-touching div expansion
        float x = p0 * rz;
        float y = p1 * rz;
        float x0f = floorf(x), y0f = floorf(y);
        float wx = x - x0f, wy = y - y0f;
        int x0 = (int)fminf(fmaxf(x0f,        0.0f), (float)(W_ - 1));
        int x1 = (int)fminf(fmaxf(x0f + 1.0f, 0.0f), (float)(W_ - 1));
        int y0 = (int)fminf(fmaxf(y0f,        0.0f), (float)(H_ - 1));
        int y1 = (int)fminf(fmaxf(y0f + 1.0f, 0.0f), (float)(H_ - 1));
        o00[j] = y0 * W_ + x0;  o01[j] = y0 * W_ + x1;
        o10[j] = y1 * W_ + x0;  o11[j] = y1 * W_ + x1;
        w00[j] = (1.0f - wx) * (1.0f - wy);
        w01[j] = wx * (1.0f - wy);
        w10[j] = (1.0f - wx) * wy;
        w11[j] = wx * wy;
        vld[j] = (z > 1e-3f) && (x >= 0.0f) && (x <= (float)(W_ - 1)) &&
                 (y >= 0.0f) && (y <= (float)(H_ - 1));
        basep[j] = src + (size_t)v * C_ * HW_;
    }

    float dac0 = 0.0f, nac0 = 0.0f, dac1 = 0.0f, nac1 = 0.0f;
#pragma unroll 8
    for (int c = 0; c < C_; ++c) {
        const size_t co = (size_t)c * HW_;
        const float rv = ref[co + pix] * rref;
        const float* pc0 = basep[0] + co;
        const float* pc1 = basep[1] + co;
        float wv0 = fmaf(pc0[o00[0]], w00[0],
                    fmaf(pc0[o01[0]], w01[0],
                    fmaf(pc0[o10[0]], w10[0], pc0[o11[0]] * w11[0])));
        float wv1 = fmaf(pc1[o00[1]], w00[1],
                    fmaf(pc1[o01[1]], w01[1],
                    fmaf(pc1[o10[1]], w10[1], pc1[o11[1]] * w11[1])));
        dac0 = fmaf(rv, wv0, dac0);
        nac0 = fmaf(wv0, wv0, nac0);
        dac1 = fmaf(rv, wv1, dac1);
        nac1 = fmaf(wv1, wv1, nac1);
    }

    // sim = dot / max(||w||, 1e-12), via v_sqrt_f32 + v_rcp_f32 (TRANS ops,
    // uniform control flow -> EXEC stays all-1s for the WMMA below).
    float rn0 = __builtin_amdgcn_rcpf(fmaxf(__builtin_amdgcn_sqrtf(nac0), 1e-12f));
    float rn1 = __builtin_amdgcn_rcpf(fmaxf(__builtin_amdgcn_sqrtf(nac1), 1e-12f));
    float sim0 = vld[0] ? (dac0 * rn0) : 0.0f;
    float sim1 = vld[1] ? (dac1 * rn1) : 0.0f;

    // View-mean via V_WMMA_F32_16X16X4_F32.
    v2f Amat; Amat.x = 0.25f; Amat.y = 0.25f;   // A: 16x4 all 0.25
    v2f Bmat; Bmat.x = sim0;  Bmat.y = sim1;    // B: 4x16, col=pixel, K=view
    v8f Cm = {};
    v8f Dm = __builtin_amdgcn_wmma_f32_16x16x4_f32(
        /*neg_a=*/false, Amat, /*neg_b=*/false, Bmat,
        /*c_mod=*/(short)0, Cm, /*reuse_a=*/false, /*reuse_b=*/false);

    if (lane < 16) {
        out[(size_t)d * HW_ + pix] = Dm[0];
    }
}

extern "C" void kernel_launch(void* const* d_in, const int* in_sizes, int n_in,
                              void* d_out, int out_size, void* d_ws, size_t ws_size,
                              hipStream_t stream) {
    const float* ref   = (const float*)d_in[0];   // (1,64,240,320)
    const float* src   = (const float*)d_in[1];   // (1,4,64,240,320)
    const float* Kref  = (const float*)d_in[2];   // (1,3,3)
    const float* c2wr  = (const float*)d_in[3];   // (1,4,4)
    const float* Ksrc  = (const float*)d_in[4];   // (1,4,3,3)
    const float* c2ws  = (const float*)d_in[5];   // (1,4,4,4)
    float* out = (float*)d_out;                   // cost (128*240*320) + depths (128)
    float* ws  = (float*)d_ws;

    setup_kernel<<<1, 128, 0, stream>>>(Kref, c2wr, Ksrc, c2ws, ws,
                                        out + (size_t)NUM_D * HW_);
    refnorm_kernel<<<(HW_ + 255) / 256, 256, 0, stream>>>(ref, ws + 64);
    dim3 grid(HW_ / 64, NUM_D);
    cost_kernel<<<grid, 128, 0, stream>>>(ref, src, ws, ws + 64, out);
}